// SpatinalGather_Module_24550033064496
// MI455X (gfx1250) — compile-verified
//
#include <hip/hip_runtime.h>
#include <hip/hip_bf16.h>
#include <math.h>

typedef __attribute__((ext_vector_type(2))) float v2f;
typedef __attribute__((ext_vector_type(8))) float v8f;

#define S_DIM 16384   // 128*128
#define C_DIM 512
#define K_DIM 19
#define B_DIM 8
#define WAVES 16      // waves per workgroup in GEMM kernel
#define SCHUNK (S_DIM / WAVES)   // 1024 s-elements per wave

// ---------------------------------------------------------------------------
// Kernel 1: row softmax over the spatial dimension.
// grid = B*K rows, block = 256 threads. Row r = b*19+k is contiguous in both
// aux (input) and probs (scratch) since aux is [B,K,H,W].
// ---------------------------------------------------------------------------
__global__ void softmax_rows(const float* __restrict__ aux,
                             float* __restrict__ probs) {
    const float* x = aux   + (size_t)blockIdx.x * S_DIM;
    float*       p = probs + (size_t)blockIdx.x * S_DIM;
    __shared__ float red[256];
    const int t = threadIdx.x;

    // pass 1: max
    float m = -INFINITY;
    for (int i = t; i < S_DIM; i += 256) m = fmaxf(m, x[i]);
    red[t] = m; __syncthreads();
    for (int off = 128; off > 0; off >>= 1) {
        if (t < off) red[t] = fmaxf(red[t], red[t + off]);
        __syncthreads();
    }
    m = red[0]; __syncthreads();

    // pass 2: sum of exp
    float s = 0.f;
    for (int i = t; i < S_DIM; i += 256) s += __expf(x[i] - m);
    red[t] = s; __syncthreads();
    for (int off = 128; off > 0; off >>= 1) {
        if (t < off) red[t] += red[t + off];
        __syncthreads();
    }
    const float inv = 1.0f / red[0];

    // pass 3: normalized probabilities
    for (int i = t; i < S_DIM; i += 256) p[i] = __expf(x[i] - m) * inv;
}

// ---------------------------------------------------------------------------
// Kernel 2: ctx[b,c,k] = sum_s feats[b,c,s] * probs[b,k,s] via f32 WMMA.
// grid = (C/16 m-tiles, B), block = 512 threads = 16 waves.
// Each wave: one 16x16 m-tile x two n-tiles (k 0..15, 16..31) over a 1024-wide
// S chunk, using V_WMMA_F32_16X16X4_F32. A fragment (feats) is shared by both
// n-tiles so feats streams from HBM exactly once — loaded non-temporally so the
// 268 MB one-shot stream does not evict the heavily reused 10 MB probs from
// L2/WGP$. Partial tiles from the 16 waves are reduced through LDS; k<19 stored.
// ---------------------------------------------------------------------------
__global__ __launch_bounds__(512)
void wmma_gemm(const float* __restrict__ feats,
               const float* __restrict__ probs,
               float* __restrict__ out) {
    const int b     = blockIdx.y;
    const int mtile = blockIdx.x;            // 0..31
    const int wave  = threadIdx.x >> 5;      // 0..15
    const int lane  = threadIdx.x & 31;
    const int row   = lane & 15;             // matrix row (A: M, B: N=k)
    const int khalf = lane >> 4;             // 0 -> K=0,1 ; 1 -> K=2,3

    // Per-lane base pointers (ISA layout for 32-bit 16x4 A / 4x16 B):
    // lane L supplies elements s0 + 2*khalf + {0,1} of its row.
    const float* Ap  = feats + ((size_t)b * C_DIM + (size_t)mtile * 16 + row) * S_DIM;
    const float* B0p = probs + ((size_t)b * K_DIM + row) * S_DIM;       // k = row (0..15 < 19)
    const int  k1    = 16 + row;
    const bool k1ok  = (k1 < K_DIM);
    const float* B1p = probs + ((size_t)b * K_DIM + (k1ok ? k1 : 0)) * S_DIM;

    int s0 = wave * SCHUNK + khalf * 2;

    v8f acc0 = {};  // k columns 0..15
    v8f acc1 = {};  // k columns 16..31 (19..31 accumulate exact zeros)

    #pragma unroll 8
    for (int it = 0; it < SCHUNK / 4; ++it, s0 += 4) {
        // A (feats): streamed once -> non-temporal load (TH=NT)
        v2f a  = __builtin_nontemporal_load((const v2f*)(Ap + s0));
        // B (probs): reused by all 32 m-tile blocks -> default (RT) caching
        v2f b0 = *(const v2f*)(B0p + s0);
        v2f b1 = *(const v2f*)(B1p + s0);
        b1.x = k1ok ? b1.x : 0.0f;           // v_cndmask, no EXEC divergence
        b1.y = k1ok ? b1.y : 0.0f;
        acc0 = __builtin_amdgcn_wmma_f32_16x16x4_f32(
                   false, a, false, b0, (short)0, acc0, false, false);
        acc1 = __builtin_amdgcn_wmma_f32_16x16x4_f32(
                   false, a, false, b1, (short)0, acc1, false, false);
    }

    // Cross-wave reduction of the 16x32 output tile through LDS.
    __shared__ float red[WAVES * 512];       // 32 KB
    float* mine = red + wave * 512;
    #pragma unroll
    for (int r = 0; r < 8; ++r) {
        mine[r * 32 + lane]       = acc0[r];
        mine[256 + r * 32 + lane] = acc1[r];
    }
    __syncthreads();

    // One thread per output element: sum 16 partials, scatter-store.
    float sum = 0.f;
    #pragma unroll
    for (int w = 0; w < WAVES; ++w) sum += red[w * 512 + threadIdx.x];

    const int tile = threadIdx.x >> 8;                 // n-tile 0/1
    const int r    = (threadIdx.x >> 5) & 7;           // accumulator VGPR index
    const int l    = threadIdx.x & 31;
    const int M    = r + ((l >= 16) ? 8 : 0);          // C/D layout: lanes 16..31 hold M+8
    const int N    = tile * 16 + (l & 15);             // k column
    const int c    = mtile * 16 + M;
    if (N < K_DIM)
        out[((size_t)b * C_DIM + c) * K_DIM + N] = sum;
}

// ---------------------------------------------------------------------------
extern "C" void kernel_launch(void* const* d_in, const int* in_sizes, int n_in,
                              void* d_out, int out_size, void* d_ws, size_t ws_size,
                              hipStream_t stream) {
    const float* bb_feats = (const float*)d_in[0];   // [8, 512, 128, 128] f32
    const float* aux_out  = (const float*)d_in[1];   // [8, 19, 128, 128]  f32
    float* out   = (float*)d_out;                    // [8, 512, 19, 1]    f32
    float* probs = (float*)d_ws;                     // [8, 19, 16384] f32 ~ 10 MB

    // 1) softmax over spatial dim, one block per (b,k) row
    softmax_rows<<<dim3(B_DIM * K_DIM), dim3(256), 0, stream>>>(aux_out, probs);

    // 2) WMMA contraction: 32 m-tiles x 8 batches, 16 waves each
    wmma_gemm<<<dim3(C_DIM / 16, B_DIM), dim3(512), 0, stream>>>(bb_feats, probs, out);
}